// GCN_27315992003070
// MI455X (gfx1250) — compile-verified
//
#include <hip/hip_runtime.h>

typedef __attribute__((ext_vector_type(2))) float v2f;
typedef __attribute__((ext_vector_type(8))) float v8f;

#define N_NODES 100000
#define N_EDGES 1600000
#define IN_DIM 512
#define HIDDEN 64
#define OUT_DIM 40

// ---------------------------------------------------------------------------
// GEMM1: H0 = X @ W1   [100000x512] x [512x64]  (memory bound: ~230MB traffic)
// Block = 256 thr (8 waves). Block tile = 32 rows x 64 cols.
// wave w: m_tile = w>>2 (0..1), n_tile = w&3 (0..3). K staged in LDS, 32/chunk.
// fp32 WMMA keeps reference precision; GEMM is HBM-bound anyway.
// ---------------------------------------------------------------------------
__global__ __launch_bounds__(256)
void gcn_gemm1(const float* __restrict__ X, const float* __restrict__ W1,
               float* __restrict__ H0) {
  __shared__ float As[32][36];   // +4 pad: 36*r mod 64 distinct for r=0..15 -> no bank conflicts
  __shared__ float Bs[32][64];
  const int t    = threadIdx.x;
  const int wave = t >> 5;
  const int lane = t & 31;
  const int half = lane >> 4;    // which 16-lane half
  const int r    = lane & 15;
  const int m0   = blockIdx.x * 32;          // 100000 % 32 == 0 -> no guard
  const int mt   = (wave >> 2) * 16;
  const int nt   = (wave & 3) * 16;

  v8f acc = {};
  for (int k0 = 0; k0 < IN_DIM; k0 += 32) {
    // cooperative, fully-coalesced float4 tile loads
    {
      const int ar = t >> 3, ac = (t & 7) << 2;              // 32 rows x 32 cols of X
      *(float4*)&As[ar][ac] =
          *(const float4*)&X[(size_t)(m0 + ar) * IN_DIM + k0 + ac];
#pragma unroll
      for (int i = 0; i < 2; ++i) {                          // 32 rows x 64 cols of W1
        const int lin = t + i * 256;
        const int br = lin >> 4, bc = (lin & 15) << 2;
        *(float4*)&Bs[br][bc] =
            *(const float4*)&W1[(size_t)(k0 + br) * HIDDEN + bc];
      }
    }
    __syncthreads();
#pragma unroll
    for (int kk = 0; kk < 32; kk += 4) {
      v2f a, b;
      // A 16x4 f32 frag: lanes 0-15 hold K=kk,kk+1 ; lanes 16-31 hold K=kk+2,kk+3
      a.x = As[mt + r][kk + 2 * half + 0];
      a.y = As[mt + r][kk + 2 * half + 1];
      // B 4x16 f32 frag: symmetric, N along lanes
      b.x = Bs[kk + 2 * half + 0][nt + r];
      b.y = Bs[kk + 2 * half + 1][nt + r];
      acc = __builtin_amdgcn_wmma_f32_16x16x4_f32(false, a, false, b,
                                                  (short)0, acc, false, false);
    }
    __syncthreads();
  }
  // C/D layout: VGPR v -> row v (lanes 0-15) / v+8 (lanes 16-31), col = r
#pragma unroll
  for (int v = 0; v < 8; ++v) {
    const int row = m0 + mt + v + 8 * half;
    H0[(size_t)row * HIDDEN + nt + r] = acc[v];
  }
}

// ---------------------------------------------------------------------------
// SpMM layer 1: H1[dst] += w * H0[src]   (64 features; one wave per edge,
// each lane owns a float2 -> 256B contiguous gather; atomics land in L2,
// the 25.6MB accumulator is fully L2-resident on a 192MB L2).
// ---------------------------------------------------------------------------
__global__ __launch_bounds__(256)
void gcn_spmm1(const int* __restrict__ esrc, const int* __restrict__ edst,
               const float* __restrict__ ew, const float* __restrict__ H0,
               float* __restrict__ H1) {
  const int wave = threadIdx.x >> 5;
  const int lane = threadIdx.x & 31;
  const int e = blockIdx.x * 8 + wave;       // 1600000 % 8 == 0 -> no guard
  const int s = esrc[e];
  const int d = edst[e];
  const float wv = ew[e];
  const float2 h = *(const float2*)&H0[(size_t)s * HIDDEN + 2 * lane];
  float* o = &H1[(size_t)d * HIDDEN + 2 * lane];
  atomicAdd(o + 0, wv * h.x);
  atomicAdd(o + 1, wv * h.y);
}

// ---------------------------------------------------------------------------
// GEMM2: H2 = relu(H1) @ W2   [100000x64] x [64x40]; ReLU fused into A-frag
// load. N padded to 48 (3 n-tiles) via zero-padded LDS copy of W2; stores
// guarded at col<40. 100000/16 = 6250 tiles; wave-uniform guard keeps
// EXEC all-ones for WMMA.
// ---------------------------------------------------------------------------
__global__ __launch_bounds__(256)
void gcn_gemm2(const float* __restrict__ H1, const float* __restrict__ W2,
               float* __restrict__ H2) {
  __shared__ float Ws[HIDDEN][48];
  const int t = threadIdx.x;
#pragma unroll
  for (int i = 0; i < 12; ++i) {             // 64x48 = 3072 = 256*12
    const int lin = t + i * 256;
    const int rr = lin / 48, cc = lin % 48;
    Ws[rr][cc] = (cc < OUT_DIM) ? W2[rr * OUT_DIM + cc] : 0.0f;
  }
  __syncthreads();

  const int wave = t >> 5;
  const int lane = t & 31;
  const int half = lane >> 4;
  const int r    = lane & 15;
  const int tile = blockIdx.x * 8 + wave;    // 16-row tile
  if (tile >= N_NODES / 16) return;          // uniform per-wave -> EXEC stays all-1
  const int m0 = tile * 16;

  v8f acc0 = {}, acc1 = {}, acc2 = {};
#pragma unroll
  for (int kk = 0; kk < HIDDEN; kk += 4) {
    v2f a;
    const float a0 = H1[(size_t)(m0 + r) * HIDDEN + kk + 2 * half + 0];
    const float a1 = H1[(size_t)(m0 + r) * HIDDEN + kk + 2 * half + 1];
    a.x = fmaxf(a0, 0.0f);                   // fused ReLU of layer-1 activation
    a.y = fmaxf(a1, 0.0f);
    v2f b0, b1, b2;
    b0.x = Ws[kk + 2 * half + 0][ 0 + r];  b0.y = Ws[kk + 2 * half + 1][ 0 + r];
    b1.x = Ws[kk + 2 * half + 0][16 + r];  b1.y = Ws[kk + 2 * half + 1][16 + r];
    b2.x = Ws[kk + 2 * half + 0][32 + r];  b2.y = Ws[kk + 2 * half + 1][32 + r];
    acc0 = __builtin_amdgcn_wmma_f32_16x16x4_f32(false, a, false, b0, (short)0, acc0, false, false);
    acc1 = __builtin_amdgcn_wmma_f32_16x16x4_f32(false, a, false, b1, (short)0, acc1, false, false);
    acc2 = __builtin_amdgcn_wmma_f32_16x16x4_f32(false, a, false, b2, (short)0, acc2, false, false);
  }
#pragma unroll
  for (int v = 0; v < 8; ++v) {
    const int row = m0 + v + 8 * half;
    H2[(size_t)row * OUT_DIM +  0 + r] = acc0[v];
    H2[(size_t)row * OUT_DIM + 16 + r] = acc1[v];
    if (r < 8) H2[(size_t)row * OUT_DIM + 32 + r] = acc2[v];
  }
}

// ---------------------------------------------------------------------------
// SpMM layer 2: out[dst] += w * H2[src]   (40 features; lanes 0..19 own a
// float2). d_out is the accumulator (zeroed each launch); final ReLU pass.
// ---------------------------------------------------------------------------
__global__ __launch_bounds__(256)
void gcn_spmm2(const int* __restrict__ esrc, const int* __restrict__ edst,
               const float* __restrict__ ew, const float* __restrict__ H2,
               float* __restrict__ O) {
  const int wave = threadIdx.x >> 5;
  const int lane = threadIdx.x & 31;
  const int e = blockIdx.x * 8 + wave;
  const int s = esrc[e];
  const int d = edst[e];
  const float wv = ew[e];
  if (lane < 20) {
    const float2 h = *(const float2*)&H2[(size_t)s * OUT_DIM + 2 * lane];
    float* o = &O[(size_t)d * OUT_DIM + 2 * lane];
    atomicAdd(o + 0, wv * h.x);
    atomicAdd(o + 1, wv * h.y);
  }
}

__global__ __launch_bounds__(256)
void gcn_relu(float* __restrict__ O, int n) {
  const int i = blockIdx.x * 256 + threadIdx.x;
  if (i < n) O[i] = fmaxf(O[i], 0.0f);
}

// ---------------------------------------------------------------------------
// kernel_launch: inputs (x, edge_src, edge_dst, edge_w, W1, W2)
// ws layout: H0 [N*64] | H1 [N*64] | H2 [N*40]  = 67.2 MB total
// ---------------------------------------------------------------------------
extern "C" void kernel_launch(void* const* d_in, const int* in_sizes, int n_in,
                              void* d_out, int out_size, void* d_ws, size_t ws_size,
                              hipStream_t stream) {
  const float* X  = (const float*)d_in[0];
  const int* esrc = (const int*)d_in[1];
  const int* edst = (const int*)d_in[2];
  const float* ew = (const float*)d_in[3];
  const float* W1 = (const float*)d_in[4];
  const float* W2 = (const float*)d_in[5];
  float* out = (float*)d_out;

  float* H0 = (float*)d_ws;
  float* H1 = H0 + (size_t)N_NODES * HIDDEN;
  float* H2 = H1 + (size_t)N_NODES * HIDDEN;

  hipMemsetAsync(H1, 0, (size_t)N_NODES * HIDDEN * sizeof(float), stream);
  hipMemsetAsync(out, 0, (size_t)N_NODES * OUT_DIM * sizeof(float), stream);

  gcn_gemm1<<<N_NODES / 32, 256, 0, stream>>>(X, W1, H0);
  gcn_spmm1<<<N_EDGES / 8, 256, 0, stream>>>(esrc, edst, ew, H0, H1);
  gcn_gemm2<<<(N_NODES / 16 + 7) / 8, 256, 0, stream>>>(H1, W2, H2);
  gcn_spmm2<<<N_EDGES / 8, 256, 0, stream>>>(esrc, edst, ew, H2, out);
  gcn_relu<<<(N_NODES * OUT_DIM + 255) / 256, 256, 0, stream>>>(
      out, N_NODES * OUT_DIM);
}